// DynamicNeuralTuringMachineMemory_37022618091851
// MI455X (gfx1250) — compile-verified
//
#include <hip/hip_runtime.h>
#include <hip/hip_bf16.h>
#include <math.h>

// ---------------------------------------------------------------------------
// DynamicNeuralTuringMachineMemory — fused, HBM-bound pipeline for MI455X
// (gfx1250, wave32). All big streams pass through V_WMMA_F32_16X16X4_F32 so
// the matrix units do the MACs while we drive ~350MB of HBM traffic once.
// ---------------------------------------------------------------------------

#define NROW   32768
#define CDIM   512
#define ADIM   64
#define CIDIM  1024
#define QROWS  576          // CDIM + ADIM
#define TOTROW 1600         // 576 (W_query) + 512 (W_erase) + 512 (W_content_hidden)
#define EPSV   1e-7f

typedef float v2f __attribute__((ext_vector_type(2)));
typedef float v8f __attribute__((ext_vector_type(8)));

// ------------------------- workspace layout (floats) -----------------------
#define WS_ACCA   0          // 1600 : fused GEMV accumulators (atomicAdd)
#define WS_S1     1600       // 512  : sum_i e_i   * mc[i,c]
#define WS_S2     2112       // 512  : sum_i e_i^2 * mc[i,c]
#define WS_TMA    2624       // 64   : sum_i e_i   * ma[i,j]
#define WS_SCAL   2688       // 16   : scalars (below)
#define WS_ZEROF  2704       // everything above gets memset to 0
#define WS_WCIX   2704       // 512  : W_content_input @ x
#define WS_QRY    3216       // 576  : query (+bias)
#define WS_ERS    3792       // 512  : erase
#define WS_CND    4304       // 512  : cand
#define WS_LGT    4816       // 32768: logits, overwritten in place by e_i

#define SC_DOT_SH 0
#define SC_DOT_LRU 1
#define SC_DOT_AH 2
#define SC_DOT_AX 3
#define SC_QNORM 4
#define SC_BETA  5
#define SC_GAMMA 6
#define SC_ALPHA 7
#define SC_MAXU  8          // ordered-uint max of logits
#define SC_Z     9          // sum e
#define SC_S3    10         // sum e^2

// ---------------------------------------------------------------------------
__device__ __forceinline__ v8f wmma_f32_k4(v2f a, v2f b, v8f c) {
  // D(16x16,f32) = A(16x4,f32) x B(4x16,f32) + C
  return __builtin_amdgcn_wmma_f32_16x16x4_f32(false, a, false, b, (short)0, c,
                                               false, false);
}

__device__ __forceinline__ float waveSum(float v) {
#pragma unroll
  for (int off = 16; off > 0; off >>= 1) v += __shfl_xor(v, off, 32);
  return v;
}
__device__ __forceinline__ float waveMax(float v) {
#pragma unroll
  for (int off = 16; off > 0; off >>= 1) v = fmaxf(v, __shfl_xor(v, off, 32));
  return v;
}
__device__ __forceinline__ unsigned mapOrdered(float f) {
  unsigned u = __float_as_uint(f);
  return (u & 0x80000000u) ? ~u : (u | 0x80000000u);
}
__device__ __forceinline__ float unmapOrdered(unsigned m) {
  unsigned u = (m & 0x80000000u) ? (m & 0x7FFFFFFFu) : ~m;
  return __uint_as_float(u);
}

// ---------------------------------------------------------------------------
// K1: fused GEMV  acc[r] = W_all[r,:] @ h   for r in [0,1600)  (209.7 MB read)
// WMMA mapping (ISA 7.12.2, 32-bit layouts):
//   A 16x4 : lane L holds row M=L&15; K slot = (L>>4)*2 + vgpr  -> float2 load
//   B 4x16 : lanes 0-15 hold K=0..1, lanes 16-31 hold K=2..3, vgpr = K within
//            we replicate h[k] across all 16 N columns -> D uniform in N.
// grid = (100 row-tiles, 64 K-chunks of 512), block = 1 wave.
// ---------------------------------------------------------------------------
__global__ void k_gemv_shared_h(const float* __restrict__ Wq,
                                const float* __restrict__ We,
                                const float* __restrict__ Wch,
                                const float* __restrict__ h,
                                float* __restrict__ ws) {
  const int rowBase = blockIdx.x * 16;
  const float* W;
  int localRow;
  if (rowBase < QROWS)            { W = Wq;  localRow = rowBase; }
  else if (rowBase < QROWS + CDIM){ W = We;  localRow = rowBase - QROWS; }
  else                            { W = Wch; localRow = rowBase - (QROWS + CDIM); }

  const int lane = threadIdx.x;
  const int m    = lane & 15;
  const int koff = (lane >> 4) * 2;
  const float* rowPtr = W + (size_t)(localRow + m) * NROW + koff;

  const int k0 = blockIdx.y * 512;
  v8f d = {};
#pragma unroll 4
  for (int kb = k0; kb < k0 + 512; kb += 4) {
    v2f a = *(const v2f*)(rowPtr + kb);          // W[row, kb+koff .. +1]
    v2f b = *(const v2f*)(h + kb + koff);        // h[kb+koff .. +1]
    d = wmma_f32_k4(a, b, d);
  }
  // D[M,N] identical over N; lanes with N==0 (lanes 0 and 16) own M = v+8*(L>>4)
  if (m == 0) {
    const int mh = (lane >> 4) * 8;
#pragma unroll
    for (int v = 0; v < 8; ++v)
      atomicAdd(&ws[WS_ACCA + rowBase + mh + v], d[v]);
  }
}

// ---------------------------------------------------------------------------
// K2: wcix[r] = W_content_input[r,:] @ x   (2 MB) — wave per row
// ---------------------------------------------------------------------------
__global__ void k_wcix(const float* __restrict__ Wci, const float* __restrict__ x,
                       float* __restrict__ ws) {
  const int row  = blockIdx.x * 4 + (threadIdx.x >> 5);
  const int lane = threadIdx.x & 31;
  float acc = 0.f;
  for (int k = lane; k < CIDIM; k += 32)
    acc += Wci[(size_t)row * CIDIM + k] * x[k];
  acc = waveSum(acc);
  if (lane == 0) ws[WS_WCIX + row] = acc;
}

// ---------------------------------------------------------------------------
// K3: scalar dot products (u_sharpen·h, u_lru·h, uca[:N]·h, uca[N:]·x)
// ---------------------------------------------------------------------------
__global__ void k_dots(const float* __restrict__ h, const float* __restrict__ x,
                       const float* __restrict__ us, const float* __restrict__ ul,
                       const float* __restrict__ uca, float* __restrict__ ws) {
  const int i = blockIdx.x * blockDim.x + threadIdx.x;   // exactly [0,32768)
  const float hv = h[i];
  float p0 = us[i] * hv;
  float p1 = ul[i] * hv;
  float p2 = uca[i] * hv;
  float p3 = (i < CIDIM) ? uca[NROW + i] * x[i] : 0.f;
  p0 = waveSum(p0); p1 = waveSum(p1); p2 = waveSum(p2); p3 = waveSum(p3);
  __shared__ float s[4];
  if (threadIdx.x < 4) s[threadIdx.x] = 0.f;
  __syncthreads();
  if ((threadIdx.x & 31) == 0) {
    atomicAdd(&s[0], p0); atomicAdd(&s[1], p1);
    atomicAdd(&s[2], p2); atomicAdd(&s[3], p3);
  }
  __syncthreads();
  if (threadIdx.x == 0) {
    atomicAdd(&ws[WS_SCAL + SC_DOT_SH],  s[0]);
    atomicAdd(&ws[WS_SCAL + SC_DOT_LRU], s[1]);
    atomicAdd(&ws[WS_SCAL + SC_DOT_AH],  s[2]);
    atomicAdd(&ws[WS_SCAL + SC_DOT_AX],  s[3]);
  }
}

// ---------------------------------------------------------------------------
// K4: finisher — query(+bias), q_norm, beta/gamma/alpha, erase, cand
// ---------------------------------------------------------------------------
__global__ void k_finish(const float* __restrict__ bq, const float* __restrict__ bsh,
                         const float* __restrict__ blru, const float* __restrict__ ber,
                         const float* __restrict__ bca, float* __restrict__ ws) {
  __shared__ float sh_q2, sh_alpha;
  const int t = threadIdx.x;
  if (t == 0) sh_q2 = 0.f;
  __syncthreads();
  if (t < QROWS) {
    float qv = ws[WS_ACCA + t] + bq[t];
    ws[WS_QRY + t] = qv;
    atomicAdd(&sh_q2, qv * qv);
  }
  __syncthreads();
  if (t == 0) {
    ws[WS_SCAL + SC_QNORM] = fmaxf(sqrtf(sh_q2), EPSV);
    float sp = ws[WS_SCAL + SC_DOT_SH] + bsh[0];                 // softplus + 1
    ws[WS_SCAL + SC_BETA] = (sp > 20.f ? sp : log1pf(expf(sp))) + 1.0f;
    float gi = ws[WS_SCAL + SC_DOT_LRU] + blru[0];
    ws[WS_SCAL + SC_GAMMA] = 1.f / (1.f + expf(-gi));
    float al = ws[WS_SCAL + SC_DOT_AH] + ws[WS_SCAL + SC_DOT_AX] + bca[0];
    ws[WS_SCAL + SC_ALPHA] = al;
    sh_alpha = al;
  }
  __syncthreads();
  if (t < CDIM) {
    ws[WS_ERS + t] = ws[WS_ACCA + QROWS + t] + ber[t];
    float cv = ws[WS_ACCA + QROWS + CDIM + t] + sh_alpha * ws[WS_WCIX + t];
    ws[WS_CND + t] = fmaxf(cv, 0.f);
  }
}

// ---------------------------------------------------------------------------
// K5: per-row cosine logits over full_mem = [MA | MC]  (75.5 MB read)
// wave per row, lanes across the 576 columns.
// ---------------------------------------------------------------------------
__global__ void k_logits(const float* __restrict__ ma, const float* __restrict__ mc,
                         const float* __restrict__ ema, float* __restrict__ ws) {
  const int row  = blockIdx.x * 8 + (threadIdx.x >> 5);
  const int lane = threadIdx.x & 31;
  const float* qry = ws + WS_QRY;
  float dot = 0.f, nrm = 0.f;
#pragma unroll 2
  for (int j = lane; j < QROWS; j += 32) {       // branch uniform per iteration
    float v = (j < ADIM) ? ma[(size_t)row * ADIM + j]
                         : mc[(size_t)row * CDIM + (j - ADIM)];
    dot += v * qry[j];
    nrm += v * v;
  }
  dot = waveSum(dot);
  nrm = waveSum(nrm);
  if (lane == 0) {
    float qn    = ws[WS_SCAL + SC_QNORM];
    float beta  = ws[WS_SCAL + SC_BETA];
    float gamma = ws[WS_SCAL + SC_GAMMA];
    float mn    = fmaxf(sqrtf(nrm), EPSV);
    float cosv  = dot / (mn * qn);
    ws[WS_LGT + row] = beta * cosv - gamma * ema[row];
  }
}

// ---------------------------------------------------------------------------
// K6: global max of logits (ordered-uint atomicMax)
// ---------------------------------------------------------------------------
__global__ void k_max(float* __restrict__ ws) {
  const int i = blockIdx.x * blockDim.x + threadIdx.x;
  float v = waveMax(ws[WS_LGT + i]);
  __shared__ float s[8];
  if ((threadIdx.x & 31) == 0) s[threadIdx.x >> 5] = v;
  __syncthreads();
  if (threadIdx.x == 0) {
    float m = s[0];
#pragma unroll
    for (int w = 1; w < 8; ++w) m = fmaxf(m, s[w]);
    atomicMax((unsigned*)&ws[WS_SCAL + SC_MAXU], mapOrdered(m));
  }
}

// ---------------------------------------------------------------------------
// K7: e_i = exp(logit - max) in place; Z = sum e; S3 = sum e^2
// ---------------------------------------------------------------------------
__global__ void k_exp(float* __restrict__ ws) {
  const unsigned mu = ((const unsigned*)ws)[WS_SCAL + SC_MAXU];
  const float mx = unmapOrdered(mu);
  const int i = blockIdx.x * blockDim.x + threadIdx.x;
  float e = expf(ws[WS_LGT + i] - mx);
  ws[WS_LGT + i] = e;
  float z  = waveSum(e);
  float z2 = waveSum(e * e);
  __shared__ float sz[8], s2[8];
  if ((threadIdx.x & 31) == 0) { sz[threadIdx.x >> 5] = z; s2[threadIdx.x >> 5] = z2; }
  __syncthreads();
  if (threadIdx.x == 0) {
    float a = 0.f, b = 0.f;
#pragma unroll
    for (int w = 0; w < 8; ++w) { a += sz[w]; b += s2[w]; }
    atomicAdd(&ws[WS_SCAL + SC_Z],  a);
    atomicAdd(&ws[WS_SCAL + SC_S3], b);
  }
}

// ---------------------------------------------------------------------------
// K8: WMMA column reductions over memory_contents (64 MB read):
//     S1[c] = sum_i e_i*mc[i,c]   (A rows 0..7  carry e)
//     S2[c] = sum_i e_i^2*mc[i,c] (A rows 8..15 carry e^2)
// B 4x16 tile of mc: lanes 0-15 rows i0+{0,1}, lanes 16-31 rows i0+{2,3};
// per-lane column c0+(L&15) -> half-wave 64B coalesced loads.
// D extract: lane<16, vgpr0 -> M=0 (S1[c0+L]); lane>=16, vgpr0 -> M=8 (S2).
// grid = (32 c-tiles, 64 i-chunks of 512), block = 1 wave.
// ---------------------------------------------------------------------------
__global__ void k_smem_wmma(const float* __restrict__ mc, float* __restrict__ ws) {
  const int c0   = blockIdx.x * 16;
  const int lane = threadIdx.x;
  const int n    = lane & 15;
  const int koff = (lane >> 4) * 2;
  const bool sq  = (n >= 8);                 // A rows 8..15 use e^2
  const float* e = ws + WS_LGT;
  const int i0b  = blockIdx.y * 512;

  v8f d = {};
#pragma unroll 4
  for (int i0 = i0b; i0 < i0b + 512; i0 += 4) {
    float e0 = e[i0 + koff];
    float e1 = e[i0 + koff + 1];
    v2f a;
    a.x = sq ? e0 * e0 : e0;
    a.y = sq ? e1 * e1 : e1;
    v2f b;
    b.x = mc[(size_t)(i0 + koff)     * CDIM + c0 + n];
    b.y = mc[(size_t)(i0 + koff + 1) * CDIM + c0 + n];
    d = wmma_f32_k4(a, b, d);
  }
  if (lane < 16) atomicAdd(&ws[WS_S1 + c0 + lane],        d[0]);   // M=0 row
  else           atomicAdd(&ws[WS_S2 + c0 + (lane - 16)], d[0]);   // M=8 row
}

// ---------------------------------------------------------------------------
// K9: T[j] = sum_i e_i * ma[i,j]   (8 MB read) — 64 threads span a row
// ---------------------------------------------------------------------------
__global__ void k_tma(const float* __restrict__ ma, float* __restrict__ ws) {
  const int j  = threadIdx.x;                 // 0..63
  const int i0 = blockIdx.x * 256;
  const float* e = ws + WS_LGT;
  float acc = 0.f;
  for (int i = i0; i < i0 + 256; ++i)
    acc += e[i] * ma[(size_t)i * ADIM + j];
  atomicAdd(&ws[WS_TMA + j], acc);
}

// ---------------------------------------------------------------------------
// K10: final read vector (576):
//  j<64 : T[j]/Z
//  j>=64: S1/Z - erase*S2/Z^2 + cand*S3/Z^2   (== new_full.T @ softmax)
// ---------------------------------------------------------------------------
__global__ void k_final(const float* __restrict__ ws, float* __restrict__ out) {
  const int t = threadIdx.x;
  const float Z    = ws[WS_SCAL + SC_Z];
  const float S3   = ws[WS_SCAL + SC_S3];
  const float invZ = 1.f / Z;
  if (t < ADIM) {
    out[t] = ws[WS_TMA + t] * invZ;
  } else if (t < QROWS) {
    const int c = t - ADIM;
    out[t] = ws[WS_S1 + c] * invZ
           - ws[WS_ERS + c] * ws[WS_S2 + c] * invZ * invZ
           + ws[WS_CND + c] * S3 * invZ * invZ;
  }
}

// ---------------------------------------------------------------------------
extern "C" void kernel_launch(void* const* d_in, const int* in_sizes, int n_in,
                              void* d_out, int out_size, void* d_ws, size_t ws_size,
                              hipStream_t stream) {
  (void)in_sizes; (void)n_in; (void)out_size; (void)ws_size;
  const float* h   = (const float*)d_in[0];
  const float* x   = (const float*)d_in[1];
  const float* mc  = (const float*)d_in[2];
  const float* ma  = (const float*)d_in[3];
  const float* ema = (const float*)d_in[4];
  const float* Wq  = (const float*)d_in[5];
  const float* bq  = (const float*)d_in[6];
  const float* ush = (const float*)d_in[7];
  const float* bsh = (const float*)d_in[8];
  const float* ulr = (const float*)d_in[9];
  const float* blr = (const float*)d_in[10];
  const float* We  = (const float*)d_in[11];
  const float* ber = (const float*)d_in[12];
  const float* Wch = (const float*)d_in[13];
  const float* Wci = (const float*)d_in[14];
  const float* uca = (const float*)d_in[15];
  const float* bca = (const float*)d_in[16];
  float* ws  = (float*)d_ws;
  float* out = (float*)d_out;

  hipMemsetAsync(d_ws, 0, WS_ZEROF * sizeof(float), stream);        // capture-safe
  k_gemv_shared_h<<<dim3(TOTROW / 16, 64), 32, 0, stream>>>(Wq, We, Wch, h, ws);
  k_wcix        <<<dim3(CDIM / 4), 128, 0, stream>>>(Wci, x, ws);
  k_dots        <<<dim3(NROW / 256), 256, 0, stream>>>(h, x, ush, ulr, uca, ws);
  k_finish      <<<1, QROWS, 0, stream>>>(bq, bsh, blr, ber, bca, ws);
  k_logits      <<<dim3(NROW / 8), 256, 0, stream>>>(ma, mc, ema, ws);
  k_max         <<<dim3(NROW / 256), 256, 0, stream>>>(ws);
  k_exp         <<<dim3(NROW / 256), 256, 0, stream>>>(ws);
  k_smem_wmma   <<<dim3(CDIM / 16, 64), 32, 0, stream>>>(mc, ws);
  k_tma         <<<dim3(NROW / 256), 64, 0, stream>>>(ma, ws);
  k_final       <<<1, QROWS, 0, stream>>>(ws, out);
}